// GDML_14955076124779
// MI455X (gfx1250) — compile-verified
//
#include <hip/hip_runtime.h>
#include <math.h>

typedef __attribute__((ext_vector_type(2))) float v2f;
typedef __attribute__((ext_vector_type(8))) float v8f;

#define N_ATOMS 21
#define DESC    210
#define DP      224          // padded descriptor length (multiple of 16)
#define M_TRAIN 12000
#define BATCH   32

#define QF      0.11180339887498949f   // sqrt(5)/20
#define ACOEF   0.0041666666666666667f // 5/(3*20^2)
#define STDV    1.0f
#define CC      0.0f

// workspace layout (in floats)
#define OFF_XS  0                        // BATCH*DP       = 7168
#define OFF_NX  (OFF_XS + BATCH*DP)      // BATCH          = 32
#define OFF_T   (OFF_NX + BATCH)         // M_TRAIN
#define OFF_S   (OFF_T + M_TRAIN)        // M_TRAIN
#define OFF_W1  (OFF_S + M_TRAIN)        // BATCH*M_TRAIN
#define OFF_W2  (OFF_W1 + BATCH*M_TRAIN) // BATCH*M_TRAIN
#define OFF_S1  (OFF_W2 + BATCH*M_TRAIN) // BATCH
#define OFF_ES  (OFF_S1 + BATCH)         // BATCH
#define OFF_F1  (OFF_ES + BATCH)         // BATCH*DP
#define OFF_F2  (OFF_F1 + BATCH*DP)      // BATCH*DP
#define NZERO   (2*BATCH + 2*BATCH*DP)   // S1, Esum, F1, F2 are contiguous

// ---------------------------------------------------------------- zero accums
__global__ void __launch_bounds__(256) kzero(float* ws) {
    float* p = ws + OFF_S1;
    for (int i = blockIdx.x * 256 + threadIdx.x; i < NZERO; i += gridDim.x * 256)
        p[i] = 0.0f;
}

// ------------------------------------------------- descriptors xs[b,d], nx[b]
__global__ void __launch_bounds__(64) kdesc(const float* __restrict__ Rs, float* ws) {
    const int b   = blockIdx.x;
    const int tid = threadIdx.x;
    __shared__ float R[N_ATOMS * 3];
    __shared__ float red[64];
    if (tid < N_ATOMS * 3) R[tid] = Rs[b * N_ATOMS * 3 + tid];
    __syncthreads();

    float* xs = ws + OFF_XS + b * DP;
    float nacc = 0.0f;
    for (int d = tid; d < DP; d += 64) {
        float val = 0.0f;
        if (d < DESC) {
            // decode tril index d -> (i,j), i>j, d = i*(i-1)/2 + j
            int i = 1, base = 0;
            while (base + i <= d) { base += i; ++i; }
            int j = d - base;
            float dx = R[i*3+0] - R[j*3+0];
            float dy = R[i*3+1] - R[j*3+1];
            float dz = R[i*3+2] - R[j*3+2];
            float dist = sqrtf(dx*dx + dy*dy + dz*dz);
            val = 1.0f / dist;
        }
        xs[d] = val;
        nacc += val * val;
    }
    red[tid] = nacc;
    __syncthreads();
    for (int o = 32; o >= 1; o >>= 1) {
        if (tid < o) red[tid] += red[tid + o];
        __syncthreads();
    }
    if (tid == 0) ws[OFF_NX + b] = red[0];
}

// ---------------------------------------- per-m: t[m]=xt.ja, s[m]=|xt|^2
__global__ void __launch_bounds__(256) kperm(const float* __restrict__ xt,
                                             const float* __restrict__ ja,
                                             float* ws) {
    const int lane = threadIdx.x & 31;
    const int m    = blockIdx.x * 8 + (threadIdx.x >> 5);
    const float* xr = xt + (size_t)m * DESC;
    const float* jr = ja + (size_t)m * DESC;
    float tacc = 0.0f, sacc = 0.0f;
    for (int d = lane; d < DESC; d += 32) {
        float x = xr[d];
        tacc += x * jr[d];
        sacc += x * x;
    }
    for (int o = 16; o >= 1; o >>= 1) {
        tacc += __shfl_xor(tacc, o, 32);
        sacc += __shfl_xor(sacc, o, 32);
    }
    if (lane == 0) { ws[OFF_T + m] = tacc; ws[OFF_S + m] = sacc; }
}

// --------------- main: G = xs.xt^T, H = xs.ja^T via WMMA f32 16x16x4; epilogue
__global__ void __launch_bounds__(32) kmain(const float* __restrict__ xt,
                                            const float* __restrict__ ja,
                                            float* ws) {
    const int mt   = blockIdx.x;          // m-tile, 0..749
    const int bt   = blockIdx.y;          // b-tile, 0..1
    const int lane = threadIdx.x;
    const int half = lane >> 4;
    const int l15  = lane & 15;
    const int m0   = mt * 16, b0 = bt * 16;
    const int koff = half * 2;

    const float* xsrow = ws + OFF_XS + (b0 + l15) * DP;          // A row (padded)
    const float* xtrow = xt + (size_t)(m0 + l15) * DESC;          // B rows
    const float* jarow = ja + (size_t)(m0 + l15) * DESC;

    v8f cG = {}; v8f cH = {};
    for (int k = 0; k + 4 <= DESC; k += 4) {                      // k = 0..204
        v2f a, bg, bh;
        a.x  = xsrow[k + koff];     a.y  = xsrow[k + koff + 1];
        bg.x = xtrow[k + koff];     bg.y = xtrow[k + koff + 1];
        bh.x = jarow[k + koff];     bh.y = jarow[k + koff + 1];
        cG = __builtin_amdgcn_wmma_f32_16x16x4_f32(false, a, false, bg, (short)0, cG, false, false);
        cH = __builtin_amdgcn_wmma_f32_16x16x4_f32(false, a, false, bh, (short)0, cH, false, false);
    }
    {   // remainder k=208..209 (A-side pad is zero for k=210,211)
        v2f a, bg = {}, bh = {};
        a.x = xsrow[208 + koff]; a.y = xsrow[209 + koff];
        if (half == 0) {
            bg.x = xtrow[208]; bg.y = xtrow[209];
            bh.x = jarow[208]; bh.y = jarow[209];
        }
        cG = __builtin_amdgcn_wmma_f32_16x16x4_f32(false, a, false, bg, (short)0, cG, false, false);
        cH = __builtin_amdgcn_wmma_f32_16x16x4_f32(false, a, false, bh, (short)0, cH, false, false);
    }

    const int m   = m0 + l15;
    const float sm = ws[OFF_S + m];
    const float tm = ws[OFF_T + m];
    float* W1 = ws + OFF_W1;
    float* W2 = ws + OFF_W2;
    #pragma unroll
    for (int i = 0; i < 8; ++i) {
        const int b = b0 + half * 8 + i;                          // C row = i + 8*half
        const float nxb = ws[OFF_NX + b];
        float d2 = QF * QF * (nxb - 2.0f * cG[i] + sm);
        d2 = fmaxf(d2, 0.0f);
        float xd  = sqrtf(d2);
        float dot = QF * (cH[i] - tm);
        float e   = ACOEF * expf(-xd);
        float e1  = e * (1.0f + xd);
        float w1  = e * dot;
        float es  = e1 * dot;
        W1[(size_t)b * M_TRAIN + m] = w1;
        W2[(size_t)b * M_TRAIN + m] = e1;
        float rw = w1, re = es;
        for (int o = 8; o >= 1; o >>= 1) {                        // reduce over 16-lane half
            rw += __shfl_xor(rw, o, 16);
            re += __shfl_xor(re, o, 16);
        }
        if (l15 == 0) {
            atomicAdd(ws + OFF_S1 + b, rw);
            atomicAdd(ws + OFF_ES + b, re);
        }
    }
}

// ------------- second GEMMs: F1 += W1 @ xt, F2 += W2 @ ja (K split in 8)
__global__ void __launch_bounds__(32) kgemm2(const float* __restrict__ xt,
                                             const float* __restrict__ ja,
                                             float* ws) {
    const int dt    = blockIdx.x;           // d-tile, 0..13
    const int bt    = blockIdx.y;           // b-tile, 0..1
    const int kc    = blockIdx.z >> 1;      // k-chunk, 0..7
    const int which = blockIdx.z & 1;       // 0 -> F1 (W1,xt), 1 -> F2 (W2,ja)
    const int lane  = threadIdx.x;
    const int half  = lane >> 4;
    const int l15   = lane & 15;
    const int d0 = dt * 16, b0 = bt * 16;
    const int kbase = kc * (M_TRAIN / 8);

    const float* A  = ws + (which ? OFF_W2 : OFF_W1);
    const float* Bm = which ? ja : xt;
    float*       acc = ws + (which ? OFF_F2 : OFF_F1);

    const float* arow = A + (size_t)(b0 + l15) * M_TRAIN;
    const int    dcol = d0 + l15;
    const bool   dvalid = dcol < DESC;

    v8f c = {};
    for (int k = kbase; k < kbase + (M_TRAIN / 8); k += 4) {
        const int ka = k + half * 2;
        v2f a, bv = {};
        a.x = arow[ka]; a.y = arow[ka + 1];
        if (dvalid) {
            bv.x = Bm[(size_t)ka * DESC + dcol];
            bv.y = Bm[(size_t)(ka + 1) * DESC + dcol];
        }
        c = __builtin_amdgcn_wmma_f32_16x16x4_f32(false, a, false, bv, (short)0, c, false, false);
    }
    #pragma unroll
    for (int i = 0; i < 8; ++i) {
        const int b = b0 + half * 8 + i;
        atomicAdd(acc + b * DP + d0 + l15, c[i]);
    }
}

// --------------------------------- final: Es and Fs assembly (tiny per-b work)
__global__ void __launch_bounds__(64) kfinal(const float* __restrict__ Rs,
                                             const float* __restrict__ ws,
                                             float* __restrict__ out) {
    const int b   = blockIdx.x;
    const int tid = threadIdx.x;
    __shared__ float R[N_ATOMS * 3];
    __shared__ float v[DESC];
    if (tid < N_ATOMS * 3) R[tid] = Rs[b * N_ATOMS * 3 + tid];
    const float* xs = ws + OFF_XS + b * DP;
    const float* F1 = ws + OFF_F1 + b * DP;
    const float* F2 = ws + OFF_F2 + b * DP;
    const float  S1b = ws[OFF_S1 + b];
    for (int p = tid; p < DESC; p += 64)
        v[p] = STDV * (QF * xs[p] * S1b - QF * F1[p] - F2[p]);
    __syncthreads();
    if (tid == 0) out[b] = CC + STDV * (ws[OFF_ES + b] / QF);
    if (tid < N_ATOMS * 3) {
        const int j = tid / 3, cdim = tid % 3;
        float acc = 0.0f;
        for (int i = 0; i < N_ATOMS; ++i) {
            if (i == j) continue;
            const int hi = (i > j) ? i : j;
            const int lo = (i > j) ? j : i;
            const int p  = hi * (hi - 1) / 2 + lo;
            const float dx = R[i*3+0] - R[j*3+0];
            const float dy = R[i*3+1] - R[j*3+1];
            const float dz = R[i*3+2] - R[j*3+2];
            const float d2 = dx*dx + dy*dy + dz*dz;
            const float dist = sqrtf(d2);
            const float inv3 = 1.0f / (d2 * dist);
            const float diffc = R[i*3+cdim] - R[j*3+cdim];
            acc += v[p] * inv3 * diffc;
        }
        out[BATCH + b * (N_ATOMS * 3) + tid] = acc;
    }
}

extern "C" void kernel_launch(void* const* d_in, const int* in_sizes, int n_in,
                              void* d_out, int out_size, void* d_ws, size_t ws_size,
                              hipStream_t stream) {
    const float* Rs = (const float*)d_in[0];
    const float* xt = (const float*)d_in[1];   // xs_train [M, DESC]
    const float* ja = (const float*)d_in[2];   // Jx_alphas [M, DESC]
    float* ws  = (float*)d_ws;
    float* out = (float*)d_out;

    kzero <<<dim3(8),              256, 0, stream>>>(ws);
    kdesc <<<dim3(BATCH),           64, 0, stream>>>(Rs, ws);
    kperm <<<dim3(M_TRAIN / 8),    256, 0, stream>>>(xt, ja, ws);
    kmain <<<dim3(M_TRAIN / 16, 2), 32, 0, stream>>>(xt, ja, ws);
    kgemm2<<<dim3(14, 2, 16),       32, 0, stream>>>(xt, ja, ws);
    kfinal<<<dim3(BATCH),           64, 0, stream>>>(Rs, ws, out);
}